// StructuralDecoder_15607911154264
// MI455X (gfx1250) — compile-verified
//
#include <hip/hip_runtime.h>

// StructuralDecoder pipeline for MI455X (gfx1250, wave32):
//   support = X@W                (WMMA f32 16x16x4)
//   logits  = A@support + b      (dominant: 256 MB stream of A;
//                                 double-buffered LDS + async-to-LDS fill)
//   assign  = softmax(logits, axis=0)
//   out     = assign^T @ X       (WMMA f32 16x16x4, K-split + LDS reduce)

typedef float v2f __attribute__((ext_vector_type(2)));
typedef float v8f __attribute__((ext_vector_type(8)));

#define N_NODES 8192
#define D_IN    256
#define D_OUT   32

// gfx1250 async global->LDS path (guarded; falls back to sync staging).
// Probe round 2 showed the builtin exists with param0 =
// '__attribute__((__vector_size__(4*sizeof(int)))) int __device__ *'
// i.e. a non-const AS(1) pointer to a 16-byte int vector.
#if __has_builtin(__builtin_amdgcn_global_load_async_to_lds_b128)
#define HAVE_ASYNC_LDS 1
typedef int v4i_vs __attribute__((vector_size(4 * sizeof(int))));
typedef __attribute__((address_space(1))) v4i_vs* g128_t;   // global src
typedef __attribute__((address_space(3))) v4i_vs* l128_t;   // LDS dst
#endif

__device__ __forceinline__ void wait_async_all() {
#if defined(HAVE_ASYNC_LDS)
#if __has_builtin(__builtin_amdgcn_s_wait_asynccnt)
  __builtin_amdgcn_s_wait_asynccnt(0);
#else
  asm volatile("s_wait_asynccnt 0x0" ::: "memory");
#endif
#endif
}

// ---------------------------------------------------------------------------
// Kernel 1: support[8192,32] = X[8192,256] @ W[256,32]
// One wave per 16-row tile. A frag: lane L holds X[r0+(L&15), kr .. kr+1].
// ---------------------------------------------------------------------------
__global__ __launch_bounds__(32) void k_support(const float* __restrict__ X,
                                                const float* __restrict__ W,
                                                float* __restrict__ S) {
  const int lane = threadIdx.x & 31;
  const int half = lane >> 4;   // 0: K pair {k,k+1}; 1: {k+2,k+3}
  const int lo   = lane & 15;
  const int r0   = blockIdx.x * 16;

  v8f acc0 = {}; v8f acc1 = {};
  const float* xrow = X + (size_t)(r0 + lo) * D_IN;

  for (int k = 0; k < D_IN; k += 4) {
    const int kr = k + 2 * half;
    v2f a = *(const v2f*)(xrow + kr);           // global b64, 8B aligned
    v2f b0, b1;
    b0.x = W[(size_t)kr       * D_OUT + lo];
    b0.y = W[(size_t)(kr + 1) * D_OUT + lo];
    b1.x = W[(size_t)kr       * D_OUT + 16 + lo];
    b1.y = W[(size_t)(kr + 1) * D_OUT + 16 + lo];
    acc0 = __builtin_amdgcn_wmma_f32_16x16x4_f32(false, a, false, b0, (short)0, acc0, false, false);
    acc1 = __builtin_amdgcn_wmma_f32_16x16x4_f32(false, a, false, b1, (short)0, acc1, false, false);
  }
  // C/D layout: lanes 0-15: VGPR r -> M=r; lanes 16-31: VGPR r -> M=8+r; N=lane&15
  for (int r = 0; r < 8; ++r) {
    const int row = r0 + r + 8 * half;
    S[(size_t)row * D_OUT + lo]      = acc0[r];
    S[(size_t)row * D_OUT + 16 + lo] = acc1[r];
  }
}

// ---------------------------------------------------------------------------
// Kernel 2 (hot): logits[8192,32] = A[8192,8192] @ support + b
// 64 threads (2 waves) per block; block owns 32 rows; K chunked by 32.
// DOUBLE-BUFFERED: prefetch chunk c+1 (async global->LDS when available)
// while computing chunk c. One barrier per chunk: the top barrier both
// publishes the prefetched tile and guarantees no wave still reads the
// buffer about to be overwritten (every wave passed compute(c) to get here).
// Pad = 36 floats keeps b128/b64 16B/8B alignment and spreads LDS banks.
// ---------------------------------------------------------------------------
#define MT  32
#define KC  32
#define PAD 36

__global__ __launch_bounds__(64) void k_assign(const float* __restrict__ A,
                                               const float* __restrict__ S,
                                               const float* __restrict__ bias,
                                               float* __restrict__ L) {
  __shared__ float At[2][MT][PAD];       // A tile   [row][k]
  __shared__ float St[2][D_OUT][PAD];    // S^T tile [n][k]

  const int tid  = threadIdx.x;
  const int lane = tid & 31;
  const int wave = tid >> 5;             // 0..1 -> 16-row subtile
  const int half = lane >> 4;
  const int lo   = lane & 15;
  const int r0   = blockIdx.x * MT;

  v8f acc0 = {}; v8f acc1 = {};

  // Stage one K-chunk into LDS buffer `buf`.
  auto stage = [&](int buf, int k0) {
    // A tile: 32x32 floats; each thread owns four float4 slots (coalesced).
    for (int i = tid; i < (MT * KC) / 4; i += 64) {
      const int f   = i * 4;
      const int row = f >> 5;            // / KC
      const int col = f & 31;
      const float* src = A + (size_t)(r0 + row) * N_NODES + k0 + col;
#if defined(HAVE_ASYNC_LDS)
      __builtin_amdgcn_global_load_async_to_lds_b128(
          (g128_t)src, (l128_t)&At[buf][row][col], 0, 0);
#else
      const float4 v = *(const float4*)src;
      At[buf][row][col + 0] = v.x;
      At[buf][row][col + 1] = v.y;
      At[buf][row][col + 2] = v.z;
      At[buf][row][col + 3] = v.w;
#endif
    }
    // Support chunk, transposed so B-frag (kr,kr+1) is one ds_load_b64.
    for (int i = tid; i < KC * D_OUT; i += 64) {
      const int kk = i >> 5;
      const int n  = i & 31;
      St[buf][n][kk] = S[(size_t)(k0 + kk) * D_OUT + n];
    }
  };

  constexpr int NCHUNK = N_NODES / KC;   // 256
  stage(0, 0);

  for (int c = 0; c < NCHUNK; ++c) {
    const int buf = c & 1;
    wait_async_all();                    // this wave's prefetch landed in LDS
    __syncthreads();                     // publish tile; protect buf^1 rewrite

    if (c + 1 < NCHUNK) stage(buf ^ 1, (c + 1) * KC);

#pragma unroll
    for (int kk = 0; kk < KC; kk += 4) {
      const int kr = kk + 2 * half;
      v2f a  = *(const v2f*)(&At[buf][wave * 16 + lo][kr]);   // ds_load_b64
      v2f b0 = *(const v2f*)(&St[buf][lo][kr]);               // {S[kr][n],S[kr+1][n]}
      v2f b1 = *(const v2f*)(&St[buf][16 + lo][kr]);
      acc0 = __builtin_amdgcn_wmma_f32_16x16x4_f32(false, a, false, b0, (short)0, acc0, false, false);
      acc1 = __builtin_amdgcn_wmma_f32_16x16x4_f32(false, a, false, b1, (short)0, acc1, false, false);
    }
  }

  const float bv0 = bias[lo];
  const float bv1 = bias[16 + lo];
  for (int r = 0; r < 8; ++r) {
    const int row = r0 + wave * 16 + r + 8 * half;
    L[(size_t)row * D_OUT + lo]      = acc0[r] + bv0;
    L[(size_t)row * D_OUT + 16 + lo] = acc1[r] + bv1;
  }
}

// ---------------------------------------------------------------------------
// Kernel 3: softmax over the NODE axis (dim 0), one block per column.
// In-place: k_assign fully rewrites L every launch, so replay is deterministic.
// ---------------------------------------------------------------------------
__global__ __launch_bounds__(256) void k_softmax(float* __restrict__ L) {
  __shared__ float red[256];
  const int c = blockIdx.x;
  const int t = threadIdx.x;

  // Pass 1: column max
  float m = -3.402823466e38f;
  for (int r = t; r < N_NODES; r += 256)
    m = fmaxf(m, L[(size_t)r * D_OUT + c]);
  red[t] = m;
  __syncthreads();
  for (int s = 128; s > 0; s >>= 1) {
    if (t < s) red[t] = fmaxf(red[t], red[t + s]);
    __syncthreads();
  }
  m = red[0];
  __syncthreads();

  // Pass 2: sum of exp
  float sum = 0.0f;
  for (int r = t; r < N_NODES; r += 256)
    sum += expf(L[(size_t)r * D_OUT + c] - m);
  red[t] = sum;
  __syncthreads();
  for (int s = 128; s > 0; s >>= 1) {
    if (t < s) red[t] += red[t + s];
    __syncthreads();
  }
  const float inv = 1.0f / red[0];

  // Pass 3: normalize (each element read+written by exactly one thread)
  for (int r = t; r < N_NODES; r += 256) {
    const size_t idx = (size_t)r * D_OUT + c;
    L[idx] = expf(L[idx] - m) * inv;
  }
}

// ---------------------------------------------------------------------------
// Kernel 4: out[32,256] = assign^T[32,8192] @ X[8192,256]
// One block per 16-col N tile; 8 waves split K=8192; LDS reduction at the end.
// ---------------------------------------------------------------------------
__global__ __launch_bounds__(256) void k_pool(const float* __restrict__ P,
                                              const float* __restrict__ X,
                                              float* __restrict__ out) {
  __shared__ float red[8][2][8][32];
  const int tid  = threadIdx.x;
  const int lane = tid & 31;
  const int wave = tid >> 5;           // 0..7 -> K split
  const int half = lane >> 4;
  const int lo   = lane & 15;
  const int n0   = blockIdx.x * 16;

  v8f acc0 = {}; v8f acc1 = {};
  const int kbeg = wave * (N_NODES / 8);
  const int kend = kbeg + (N_NODES / 8);

  for (int k = kbeg; k < kend; k += 4) {
    const int kr = k + 2 * half;
    v2f a0, a1, b;
    a0.x = P[(size_t)kr       * D_OUT + lo];        // m tile 0 (rows 0..15)
    a0.y = P[(size_t)(kr + 1) * D_OUT + lo];
    a1.x = P[(size_t)kr       * D_OUT + 16 + lo];   // m tile 1 (rows 16..31)
    a1.y = P[(size_t)(kr + 1) * D_OUT + 16 + lo];
    b.x  = X[(size_t)kr       * D_IN + n0 + lo];
    b.y  = X[(size_t)(kr + 1) * D_IN + n0 + lo];
    acc0 = __builtin_amdgcn_wmma_f32_16x16x4_f32(false, a0, false, b, (short)0, acc0, false, false);
    acc1 = __builtin_amdgcn_wmma_f32_16x16x4_f32(false, a1, false, b, (short)0, acc1, false, false);
  }

  for (int r = 0; r < 8; ++r) {
    red[wave][0][r][lane] = acc0[r];
    red[wave][1][r][lane] = acc1[r];
  }
  __syncthreads();

  // 512 output elements for this block; each thread finalizes 2.
  for (int e = tid; e < 512; e += 256) {
    const int frag = e >> 8;
    const int r    = (e >> 5) & 7;
    const int ln   = e & 31;
    float s = 0.0f;
    for (int w = 0; w < 8; ++w) s += red[w][frag][r][ln];
    const int m = frag * 16 + r + 8 * (ln >> 4);
    const int n = n0 + (ln & 15);
    out[(size_t)m * D_IN + n] = s;   // written exactly once per call
  }
}

// ---------------------------------------------------------------------------
extern "C" void kernel_launch(void* const* d_in, const int* in_sizes, int n_in,
                              void* d_out, int out_size, void* d_ws, size_t ws_size,
                              hipStream_t stream) {
  const float* X = (const float*)d_in[0];   // main_feat [8192,256]
  const float* A = (const float*)d_in[1];   // main_adj  [8192,8192]
  const float* W = (const float*)d_in[2];   // W         [256,32]
  const float* b = (const float*)d_in[3];   // b         [32]
  float* out = (float*)d_out;               // raw_emb   [32,256]

  float* S = (float*)d_ws;                  // support [8192,32]  (1 MB)
  float* L = S + (size_t)N_NODES * D_OUT;   // logits/assign      (1 MB)

  k_support<<<N_NODES / 16, 32, 0, stream>>>(X, W, S);
  k_assign <<<N_NODES / MT, 64, 0, stream>>>(A, S, b, L);
  k_softmax<<<D_OUT, 256, 0, stream>>>(L);
  k_pool   <<<D_IN / 16, 256, 0, stream>>>(L, X, out);
}